// SegmentCompressor_13580686590437
// MI455X (gfx1250) — compile-verified
//
#include <hip/hip_runtime.h>
#include <hip/hip_bf16.h>
#include <stdint.h>

// Problem geometry (from the reference): B=8, S=4096, D=512.
// S and D are hardcoded (S must be a power of two for the blockIdx split).
#define S_LEN 4096
#define S_SHIFT 12
#define D_DIM 512
#define THRESH_BITS 0.5f

// -------------------------------------------------------------------------
// CDNA5 async global->LDS copy helpers (gfx1250). Each lane copies 16 bytes
// from global into its own LDS slot; tracked with ASYNCcnt (per-wave).
// -------------------------------------------------------------------------
__device__ __forceinline__ void async_copy_b128(uint32_t lds_byte_off,
                                                const float* gptr) {
  asm volatile("global_load_async_to_lds_b128 %0, %1, off"
               :
               : "v"(lds_byte_off), "v"(gptr)
               : "memory");
}
__device__ __forceinline__ void wait_async_le1() {
  asm volatile("s_wait_asynccnt 0x1" ::: "memory");
}
__device__ __forceinline__ void wait_async_le0() {
  asm volatile("s_wait_asynccnt 0x0" ::: "memory");
}
__device__ __forceinline__ void wait_ds0() {
  asm volatile("s_wait_dscnt 0x0" ::: "memory");
}
// Low 32 bits of a generic pointer to a __shared__ object == LDS byte offset.
__device__ __forceinline__ uint32_t lds_off_of(const void* p) {
  return (uint32_t)(uintptr_t)p;
}

// -------------------------------------------------------------------------
// Kernel 1: init per-segment scratch (cnt=0, first=+inf, last=-1).
// -------------------------------------------------------------------------
__global__ void seg_init_kernel(int* __restrict__ cnt, int* __restrict__ first,
                                int* __restrict__ last, int n) {
  int i = blockIdx.x * blockDim.x + threadIdx.x;
  if (i < n) {
    cnt[i] = 0;
    first[i] = 0x7fffffff;
    last[i] = -1;
  }
}

// -------------------------------------------------------------------------
// Kernel 2: per-batch scan. One block (256 threads) per batch row of S=4096.
// Each thread owns 16 consecutive positions.
//   - block-reduce n_valid
//   - patch_end flags, block exclusive scan -> seg_id
//   - emit seg_id / patch_end outputs (as float), seg_for_pool + per-segment
//     cnt/first/last + n_patches to workspace.
// -------------------------------------------------------------------------
__global__ __launch_bounds__(256) void seg_scan_kernel(
    const float* __restrict__ entropy, const unsigned char* __restrict__ mask,
    float* __restrict__ out_seg_id, float* __restrict__ out_patch_end,
    int* __restrict__ segpool, int* __restrict__ cnt, int* __restrict__ first,
    int* __restrict__ last, int* __restrict__ npatch) {
  const int b = blockIdx.x;
  const int tid = threadIdx.x;
  const int base = b * S_LEN + tid * 16;
  __shared__ int sdata[256];

  float ev[16];
  unsigned char mv[16];
  int nv_local = 0;
#pragma unroll
  for (int i = 0; i < 16; ++i) {
    ev[i] = entropy[base + i];
    mv[i] = mask[base + i];
    nv_local += (mv[i] == 0) ? 1 : 0;
  }
  // block reduction -> n_valid
  sdata[tid] = nv_local;
  __syncthreads();
  for (int o = 128; o > 0; o >>= 1) {
    if (tid < o) sdata[tid] += sdata[tid + o];
    __syncthreads();
  }
  const int n_valid = sdata[0];
  __syncthreads();

  // patch-end flags
  int pe[16];
  int local_sum = 0;
#pragma unroll
  for (int i = 0; i < 16; ++i) {
    const int t = tid * 16 + i;
    const bool valid = (mv[i] == 0);
    const bool is_last = valid && (t == n_valid - 1);
    pe[i] = (((ev[i] > THRESH_BITS) && valid) || is_last) ? 1 : 0;
    local_sum += pe[i];
  }
  // block inclusive scan (Hillis-Steele) over per-thread sums
  sdata[tid] = local_sum;
  __syncthreads();
  for (int o = 1; o < 256; o <<= 1) {
    int add = (tid >= o) ? sdata[tid - o] : 0;
    __syncthreads();
    sdata[tid] += add;
    __syncthreads();
  }
  int running = sdata[tid] - local_sum;  // exclusive prefix for this thread
  if (tid == 255) npatch[b] = sdata[255];  // total patch ends = n_patches

#pragma unroll
  for (int i = 0; i < 16; ++i) {
    const int t = tid * 16 + i;
    const int idx = b * S_LEN + t;
    const bool valid = (mv[i] == 0);
    const int seg = running;  // exclusive cumsum => seg_id
    running += pe[i];
    out_seg_id[idx] = (float)seg;
    out_patch_end[idx] = pe[i] ? 1.0f : 0.0f;
    segpool[idx] = valid ? seg : S_LEN;
    if (valid) {
      atomicAdd(&cnt[b * S_LEN + seg], 1);
      atomicMin(&first[b * S_LEN + seg], t);
      atomicMax(&last[b * S_LEN + seg], t);
    }
  }
}

// -------------------------------------------------------------------------
// Kernel 3: segment mean pooling. One block per (b, p); 128 threads each
// handle a float4 of the D=512 row. Rows stream through an async
// global->LDS double buffer: each lane reads back exactly the 16B it
// copied, so a per-wave s_wait_asynccnt is sufficient (no block barrier).
// -------------------------------------------------------------------------
__global__ __launch_bounds__(128) void seg_pool_kernel(
    const float* __restrict__ emb, const int* __restrict__ segpool,
    const int* __restrict__ cnt, const int* __restrict__ first,
    const int* __restrict__ last, const int* __restrict__ npatch,
    float* __restrict__ out_prevq, float* __restrict__ out_first,
    float* __restrict__ out_validmask) {
  const int p = blockIdx.x & (S_LEN - 1);
  const int b = blockIdx.x >> S_SHIFT;
  const int tid = threadIdx.x;

  __shared__ __align__(16) float buf[2][D_DIM];

  const int np = npatch[b];
  float4 acc = make_float4(0.f, 0.f, 0.f, 0.f);
  int lo = 0;

  if (p < np) {
    const int sidx = b * S_LEN + p;
    lo = first[sidx];
    const int hi = last[sidx];
    const int c = cnt[sidx];
    const float* rowbase = emb + (size_t)(b * S_LEN) * D_DIM;
    const uint32_t lds0 = lds_off_of(&buf[0][tid * 4]);
    const uint32_t lds1 = lds_off_of(&buf[1][tid * 4]);

    // prologue: prefetch row lo into buffer 0
    async_copy_b128(lds0, rowbase + (size_t)lo * D_DIM + tid * 4);

    for (int t = lo; t <= hi; ++t) {
      const int k = (t - lo) & 1;
      if (t < hi) {
        wait_ds0();  // prior ds read of the buffer we are about to overwrite
        async_copy_b128(k ? lds0 : lds1,
                        rowbase + (size_t)(t + 1) * D_DIM + tid * 4);
        wait_async_le1();  // row t complete (in-order completion)
      } else {
        wait_async_le0();
      }
      const float4 v = *(const float4*)&buf[k][tid * 4];
      if (segpool[b * S_LEN + t] == p) {  // skip padded bytes inside the span
        acc.x += v.x;
        acc.y += v.y;
        acc.z += v.z;
        acc.w += v.w;
      }
    }
    const float inv = 1.0f / (float)(c > 0 ? c : 1);
    acc.x *= inv;
    acc.y *= inv;
    acc.z *= inv;
    acc.w *= inv;
  }
  // invalid patches write zeros (d_out is poisoned by the harness)
  float4* outp = (float4*)(out_prevq + (size_t)blockIdx.x * D_DIM);
  outp[tid] = acc;
  if (tid == 0) {
    out_first[blockIdx.x] = (p < np) ? (float)lo : 0.0f;
    out_validmask[blockIdx.x] = (p < np) ? 1.0f : 0.0f;
  }
}

// -------------------------------------------------------------------------
// Launch. Outputs concatenated flat: pre_vq (B*S*D), seg_id (B*S),
// first_byte_idx (B*S), valid_mask (B*S), patch_end_mask (B*S), all fp32.
// -------------------------------------------------------------------------
extern "C" void kernel_launch(void* const* d_in, const int* in_sizes, int n_in,
                              void* d_out, int out_size, void* d_ws,
                              size_t ws_size, hipStream_t stream) {
  const float* emb = (const float*)d_in[0];
  const float* ent = (const float*)d_in[1];
  const unsigned char* mask = (const unsigned char*)d_in[2];  // numpy bool

  const int BS = in_sizes[1];  // B*S
  const int B = BS / S_LEN;

  float* out = (float*)d_out;
  float* out_prevq = out;
  float* out_segid = out + (size_t)BS * D_DIM;
  float* out_first = out_segid + BS;
  float* out_valid = out_first + BS;
  float* out_pend = out_valid + BS;

  int* ws = (int*)d_ws;
  int* segpool = ws;            // BS
  int* cnt = segpool + BS;      // BS
  int* first = cnt + BS;        // BS
  int* last = first + BS;       // BS
  int* npatch = last + BS;      // B

  seg_init_kernel<<<(BS + 255) / 256, 256, 0, stream>>>(cnt, first, last, BS);
  seg_scan_kernel<<<B, 256, 0, stream>>>(ent, mask, out_segid, out_pend,
                                         segpool, cnt, first, last, npatch);
  seg_pool_kernel<<<BS, 128, 0, stream>>>(emb, segpool, cnt, first, last,
                                          npatch, out_prevq, out_first,
                                          out_valid);
}